// mask_dice_bpm_inverse_23270132809779
// MI455X (gfx1250) — compile-verified
//
#include <hip/hip_runtime.h>
#include <hip/hip_bf16.h>
#include <cstdint>
#include <cstddef>

// Problem geometry (fixed by the reference)
#define B_        2
#define L_        8
#define DIM_      128
#define HW_       (128 * 128)
#define VOL_      (128 * 128 * 128)

// Tile geometry
#define TZ        4
#define TY        16
#define TX        16
#define HZ        (TZ + 2)          // 6
#define HY        (TY + 2)          // 18
#define HX        (TX + 2)          // 18
#define HPLANE    (HY * HX)         // 324
#define HVOL      (HZ * HY * HX)    // 1944
#define HTOT      (L_ * HVOL)       // 15552 floats = 62208 B of LDS
#define NTHREADS  256
#define TILES_PER_B ((DIM_ / TZ) * (DIM_ / TY) * (DIM_ / TX))  // 32*8*8 = 2048
#define NBLOCKS   (B_ * TILES_PER_B)                            // 4096

#define CLAMP_MIN 0.071f
#define EPS_      1e-5f

// ---- CDNA5 async global->LDS path (guarded; fallback = plain ld/st) ----
#if defined(__has_builtin)
#if __has_builtin(__builtin_amdgcn_global_load_async_to_lds_b32) && \
    __has_builtin(__builtin_amdgcn_s_wait_asynccnt)
#define USE_ASYNC_LDS 1
#endif
#endif
#ifndef USE_ASYNC_LDS
#define USE_ASYNC_LDS 0
#endif

typedef __attribute__((address_space(1))) int* gint_ptr;
typedef __attribute__((address_space(3))) int* lint_ptr;

__device__ __forceinline__ int reflect128(int i) {
  // jnp.pad(..., mode='reflect') with pad=1:  -1 -> 1,  128 -> 126
  i = (i < 0) ? -i : i;
  i = (i > 127) ? (254 - i) : i;
  return i;
}

__global__ __launch_bounds__(NTHREADS)
void fused_dice_bpm_kernel(const float* __restrict__ pred,
                           const float* __restrict__ target,
                           float* __restrict__ ptop,
                           float* __restrict__ pbot) {
  __shared__ float tile[HTOT];      // raw pred halo, then (in-place) clamped softmax
  __shared__ float redA[NTHREADS];
  __shared__ float redB[NTHREADS];

  const int tid = threadIdx.x;
  const int blk = blockIdx.x;
  const int b   = blk / TILES_PER_B;
  const int t   = blk - b * TILES_PER_B;
  const int z0  = (t >> 6) * TZ;        // 64 xy-tiles per z-slab
  const int y0  = ((t >> 3) & 7) * TY;
  const int x0  = (t & 7) * TX;

  // ---- Stage 1: reflect-padded 8-label halo tile -> LDS (async copy) ----
  for (int i = tid; i < HTOT; i += NTHREADS) {
    int l  = i / HVOL;
    int r  = i - l * HVOL;
    int zz = r / HPLANE;
    int rr = r - zz * HPLANE;
    int yy = rr / HX;
    int xx = rr - yy * HX;
    int gz = reflect128(z0 + zz - 1);
    int gy = reflect128(y0 + yy - 1);
    int gx = reflect128(x0 + xx - 1);
    size_t gidx = (size_t)(b * L_ + l) * (size_t)VOL_ +
                  (size_t)gz * (size_t)HW_ + (size_t)(gy * DIM_ + gx);
#if USE_ASYNC_LDS
    __builtin_amdgcn_global_load_async_to_lds_b32(
        (gint_ptr)(pred + gidx),
        (lint_ptr)(tile + i),
        0, 0);
#else
    tile[i] = pred[gidx];
#endif
  }
#if USE_ASYNC_LDS
  __builtin_amdgcn_s_wait_asynccnt(0);   // per-wave: all my async LDS writes landed
#endif
  __syncthreads();                       // all waves' copies visible

  // ---- Stage 2: in-place 8-way softmax + clamp per padded voxel ----
  for (int p = tid; p < HVOL; p += NTHREADS) {
    float v[L_];
    float m = -3.4e38f;
#pragma unroll
    for (int l = 0; l < L_; ++l) {
      v[l] = tile[l * HVOL + p];
      m = fmaxf(m, v[l]);
    }
    float s = 0.f;
#pragma unroll
    for (int l = 0; l < L_; ++l) {
      v[l] = __expf(v[l] - m);
      s += v[l];
    }
    float inv = 1.f / s;
#pragma unroll
    for (int l = 0; l < L_; ++l) {
      tile[l * HVOL + p] = fmaxf(v[l] * inv, CLAMP_MIN);
    }
  }
  __syncthreads();

  // ---- Stage 3: row-based separable 27-tap stencil + fused dice partials ----
  // Task = one x-row (16 voxels) of one label. 512 tasks, two halves of 256.
  // half 0 -> labels 0..3, half 1 -> labels 4..7 (l = half*4 + tid/64).
  for (int half = 0; half < 2; ++half) {
    const int task = half * NTHREADS + tid;
    const int l    = task >> 6;          // label
    const int row  = task & 63;          // row within tile
    const int z    = row >> 4;           // 0..3
    const int y    = row & 15;           // 0..15
    const float* lt = tile + l * HVOL;

    float colsum[HX];
#pragma unroll
    for (int x = 0; x < HX; ++x) colsum[x] = 0.f;
    float mid[HX];

#pragma unroll
    for (int dz = 0; dz < 3; ++dz) {
#pragma unroll
      for (int dy = 0; dy < 3; ++dy) {
        const int base = (z + dz) * HPLANE + (y + dy) * HX;
#pragma unroll
        for (int x = 0; x < HX; ++x) {
          const float vv = lt[base + x];
          colsum[x] += vv;
          if (dz == 1 && dy == 1) mid[x] = vv;
        }
      }
    }

    // Raw pred / target for this output row: 16 consecutive floats (64B aligned)
    const size_t gbase = (size_t)(b * L_ + l) * (size_t)VOL_ +
                         (size_t)(z0 + z) * (size_t)HW_ +
                         (size_t)((y0 + y) * DIM_ + x0);
    float top = 0.f, bot = 0.f;
#pragma unroll
    for (int x = 0; x < TX; ++x) {
      const float s27 = colsum[x] + colsum[x + 1] + colsum[x + 2];
      const float bt  = fabsf(27.f * mid[x + 1] - s27);   // |26*c - (s27-c)|
      const float w   = 0.5f / (bt + 0.5f);
      const float yp  = pred[gbase + x];                  // dice uses RAW logits
      const float tt  = target[gbase + x];
      top += tt * yp * w;
      bot += (tt + yp) * w;
    }

    redA[tid] = top;
    redB[tid] = bot;
    __syncthreads();
    // threads [l64*64, l64*64+64) all hold the same label; 4 serial summers
    if (tid < 4) {
      const int lsum = half * 4 + tid;
      float ts = 0.f, bs = 0.f;
      for (int r = 0; r < 64; ++r) {
        ts += redA[tid * 64 + r];
        bs += redB[tid * 64 + r];
      }
      ptop[blk * L_ + lsum] = ts;
      pbot[blk * L_ + lsum] = bs;
    }
    __syncthreads();
  }
}

// ---- Final deterministic reduction: 2048 tiles per (b,l), fixed serial order ----
__global__ __launch_bounds__(32)
void finalize_loss_kernel(const float* __restrict__ ptop,
                          const float* __restrict__ pbot,
                          float* __restrict__ out) {
  __shared__ float ratio[16];
  const int bl = threadIdx.x;
  if (bl < 16) {
    const int b = bl >> 3;
    const int l = bl & 7;
    float ts = 0.f, bs = 0.f;
    for (int tTile = 0; tTile < TILES_PER_B; ++tTile) {
      const int blk = b * TILES_PER_B + tTile;
      ts += ptop[blk * L_ + l];
      bs += pbot[blk * L_ + l];
    }
    const float bottom = fmaxf(bs, EPS_);
    ratio[bl] = (2.f * ts) / bottom;
  }
  __syncthreads();
  if (threadIdx.x == 0) {
    float s = 0.f;
    for (int i = 0; i < 16; ++i) s += ratio[i];
    out[0] = -(s / 16.f);
  }
}

extern "C" void kernel_launch(void* const* d_in, const int* in_sizes, int n_in,
                              void* d_out, int out_size, void* d_ws, size_t ws_size,
                              hipStream_t stream) {
  (void)in_sizes; (void)n_in; (void)out_size; (void)ws_size;
  const float* pred   = (const float*)d_in[0];
  const float* target = (const float*)d_in[1];
  float* out  = (float*)d_out;
  float* ptop = (float*)d_ws;                       // NBLOCKS * 8 floats
  float* pbot = ptop + (size_t)NBLOCKS * L_;        // NBLOCKS * 8 floats

  fused_dice_bpm_kernel<<<NBLOCKS, NTHREADS, 0, stream>>>(pred, target, ptop, pbot);
  finalize_loss_kernel<<<1, 32, 0, stream>>>(ptop, pbot, out);
}